// LennardJonesForce_7687991460464
// MI455X (gfx1250) — compile-verified
//
#include <hip/hip_runtime.h>
#include <hip/hip_bf16.h>

typedef __attribute__((ext_vector_type(2))) float v2f;
typedef __attribute__((ext_vector_type(8))) float v8f;

#define LJ_BOX      10.0f
#define LJ_INV_BOX  0.1f
#define LJ_RC       1.0f
#define LJ_RS       0.9f
#define LJ_INV_W    (1.0f / (LJ_RC - LJ_RS))

#define NBLOCKS 1024
#define TPB     256
#define NQ      16      // 11 real quantities, padded to 16 rows for WMMA

// ---------------------------------------------------------------------------
// Zero the output buffer (forces are accumulated with atomics; scalars are
// overwritten by the reducer but zeroing everything is cheap and safe).
// ---------------------------------------------------------------------------
__global__ void lj_zero_kernel(float* __restrict__ p, int n) {
    for (int idx = blockIdx.x * blockDim.x + threadIdx.x; idx < n;
         idx += blockDim.x * gridDim.x)
        p[idx] = 0.0f;
}

// ---------------------------------------------------------------------------
// Main per-edge kernel: gather, minimum image, LJ + 5th-order switch,
// scatter forces with global f32 atomics, block-reduce 11 scalars in LDS,
// write per-block partials to ws as a quantity-major [NQ][NBLOCKS] matrix.
// ---------------------------------------------------------------------------
__global__ void lj_edge_kernel(const float* __restrict__ pos,
                               const float* __restrict__ eps,
                               const float* __restrict__ sig,
                               const long long* __restrict__ ei,
                               float* __restrict__ forces,   // d_out + 1
                               float* __restrict__ ws,       // [NQ][nb]
                               int n_edges, int nb)
{
    __shared__ float part[NQ];
    if (threadIdx.x < NQ) part[threadIdx.x] = 0.0f;
    __syncthreads();

    float e_acc = 0.0f, w_acc = 0.0f;
    float vt0 = 0.f, vt1 = 0.f, vt2 = 0.f, vt3 = 0.f, vt4 = 0.f,
          vt5 = 0.f, vt6 = 0.f, vt7 = 0.f, vt8 = 0.f;

    const int stride = blockDim.x * gridDim.x;
    for (int e = blockIdx.x * blockDim.x + threadIdx.x; e < n_edges; e += stride) {
        // Software prefetch of the streaming edge data a few iterations ahead
        // (speculative global_prefetch_b8; OOB translations are dropped).
        int pe = e + 4 * stride;
        __builtin_prefetch(&eps[pe], 0, 1);
        __builtin_prefetch(&sig[pe], 0, 1);
        __builtin_prefetch(&ei[pe], 0, 1);
        __builtin_prefetch(&ei[n_edges + pe], 0, 1);

        const int i = (int)ei[e];
        const int j = (int)ei[n_edges + e];

        // gather positions (L2-resident)
        const float pix = pos[3 * i + 0], piy = pos[3 * i + 1], piz = pos[3 * i + 2];
        const float pjx = pos[3 * j + 0], pjy = pos[3 * j + 1], pjz = pos[3 * j + 2];

        // minimum image (rintf == round-half-to-even, matches jnp.round)
        float rx = pix - pjx; rx -= LJ_BOX * rintf(rx * LJ_INV_BOX);
        float ry = piy - pjy; ry -= LJ_BOX * rintf(ry * LJ_INV_BOX);
        float rz = piz - pjz; rz -= LJ_BOX * rintf(rz * LJ_INV_BOX);

        const float r2 = fmaxf(rx * rx + ry * ry + rz * rz, 1e-24f);
        const float r = sqrtf(r2);
        const float inv_r = 1.0f / r;

        const float epsv = eps[e];
        const float sigv = sig[e];
        const float sr  = sigv * inv_r;
        const float sr2 = sr * sr;
        const float sr6 = sr2 * sr2 * sr2;
        const float sr12 = sr6 * sr6;

        const float u_raw    = 4.0f * epsv * (sr12 - sr6);
        const float fmag_raw = 24.0f * epsv * inv_r * (2.0f * sr12 - sr6);

        // 5th-order switching function on [RS, RC]
        float x = (r - LJ_RS) * LJ_INV_W;
        x = fminf(fmaxf(x, 0.0f), 1.0f);
        const float x2 = x * x, x3 = x2 * x, x4 = x2 * x2, x5 = x4 * x;
        const float s_mid  = 1.0f - 10.0f * x3 + 15.0f * x4 - 6.0f * x5;
        const float ds_mid = (-30.0f * x2 + 60.0f * x3 - 30.0f * x4) * LJ_INV_W;
        const bool  in_mid = (r > LJ_RS) && (r < LJ_RC);
        const float s  = (r <= LJ_RS) ? 1.0f : (in_mid ? s_mid : 0.0f);
        const float ds = in_mid ? ds_mid : 0.0f;

        const float fmag = fmag_raw * s - u_raw * ds;
        e_acc += u_raw * s;
        w_acc += (r < LJ_RC) ? fmag * r : 0.0f;

        const float fsc = fmag * inv_r;
        const float fx = fsc * rx, fy = fsc * ry, fz = fsc * rz;

        // scatter-add pair forces (global_atomic_add_f32, no return)
        atomicAdd(&forces[3 * i + 0], fx);
        atomicAdd(&forces[3 * i + 1], fy);
        atomicAdd(&forces[3 * i + 2], fz);
        atomicAdd(&forces[3 * j + 0], -fx);
        atomicAdd(&forces[3 * j + 1], -fy);
        atomicAdd(&forces[3 * j + 2], -fz);

        // virial tensor: vt[a][b] += f_a * r_b
        vt0 += fx * rx; vt1 += fx * ry; vt2 += fx * rz;
        vt3 += fy * rx; vt4 += fy * ry; vt5 += fy * rz;
        vt6 += fz * rx; vt7 += fz * ry; vt8 += fz * rz;
    }

    // block reduction via LDS float atomics (ds_add_f32)
    atomicAdd(&part[0], e_acc);
    atomicAdd(&part[1], w_acc);
    atomicAdd(&part[2], vt0);  atomicAdd(&part[3], vt1);  atomicAdd(&part[4], vt2);
    atomicAdd(&part[5], vt3);  atomicAdd(&part[6], vt4);  atomicAdd(&part[7], vt5);
    atomicAdd(&part[8], vt6);  atomicAdd(&part[9], vt7);  atomicAdd(&part[10], vt8);
    __syncthreads();

    // quantity-major partials; rows 11..15 are written as 0 (WMMA padding)
    if (threadIdx.x < NQ)
        ws[threadIdx.x * nb + blockIdx.x] = part[threadIdx.x];
}

// ---------------------------------------------------------------------------
// Final reduction over blocks using V_WMMA_F32_16X16X4_F32.
// ws is a 16 x nb matrix (quantity-major). We accumulate
//   D += A_chunk(16x4) * ones(4x16)
// so each row m of D becomes the exact f32-FMA sum of quantity m over all
// blocks. A-layout (16x4 f32, wave32): lanes 0-15 hold (M=lane, K=0..1) in
// VGPR0..1; lanes 16-31 hold (M=lane-16, K=2..3).
// D-layout: VGPR r, lanes 0-15 -> (M=r, N=lane); lanes 16-31 -> (M=r+8).
// Launched as exactly one wave (32 threads), EXEC all-1s.
// ---------------------------------------------------------------------------
__global__ void lj_reduce_kernel(const float* __restrict__ ws,
                                 float* __restrict__ out,
                                 int nb, int n_atoms)
{
    const int lane = threadIdx.x;
#if __has_builtin(__builtin_amdgcn_wmma_f32_16x16x4_f32)
    const int m     = (lane < 16) ? lane : (lane - 16);
    const int kbase = (lane < 16) ? 0 : 2;

    v8f acc = {};
    v2f ones; ones.x = 1.0f; ones.y = 1.0f;

    for (int c = 0; c < nb; c += 4) {
        v2f a;
        a.x = ws[m * nb + c + kbase + 0];
        a.y = ws[m * nb + c + kbase + 1];
        acc = __builtin_amdgcn_wmma_f32_16x16x4_f32(
            /*neg_a=*/false, a, /*neg_b=*/false, ones,
            /*c_mod=*/(short)0, acc, /*reuse_a=*/false, /*reuse_b=*/false);
    }

    // Column N=0 lives in lane 0 (rows 0..7) and lane 16 (rows 8..15).
    if (lane == 0) {
        out[0]               = acc[0];                 // energy
        out[1 + 3 * n_atoms] = acc[1];                 // virial
        out[2 + 3 * n_atoms + 0] = acc[2];             // tensor[0..5]
        out[2 + 3 * n_atoms + 1] = acc[3];
        out[2 + 3 * n_atoms + 2] = acc[4];
        out[2 + 3 * n_atoms + 3] = acc[5];
        out[2 + 3 * n_atoms + 4] = acc[6];
        out[2 + 3 * n_atoms + 5] = acc[7];
    } else if (lane == 16) {
        out[2 + 3 * n_atoms + 6] = acc[0];             // tensor[6..8]
        out[2 + 3 * n_atoms + 7] = acc[1];
        out[2 + 3 * n_atoms + 8] = acc[2];
    }
#else
    // Scalar fallback (keeps compile green if the builtin is absent).
    if (lane == 0) {
        for (int q = 0; q < 11; ++q) {
            float s = 0.0f;
            for (int b = 0; b < nb; ++b) s += ws[q * nb + b];
            if (q == 0)      out[0] = s;
            else if (q == 1) out[1 + 3 * n_atoms] = s;
            else             out[2 + 3 * n_atoms + (q - 2)] = s;
        }
    }
#endif
}

extern "C" void kernel_launch(void* const* d_in, const int* in_sizes, int n_in,
                              void* d_out, int out_size, void* d_ws, size_t ws_size,
                              hipStream_t stream) {
    const float*     pos = (const float*)d_in[0];
    const float*     eps = (const float*)d_in[1];
    const float*     sig = (const float*)d_in[2];
    const long long* ei  = (const long long*)d_in[3];

    const int n_atoms = in_sizes[0] / 3;
    const int n_edges = in_sizes[1];

    float* out = (float*)d_out;
    float* ws  = (float*)d_ws;     // needs NQ * NBLOCKS * 4 = 64 KB

    // 1) zero outputs (forces accumulate atomically from zero)
    lj_zero_kernel<<<(out_size + 255) / 256, 256, 0, stream>>>(out, out_size);

    // 2) main edge loop; forces live at out+1
    lj_edge_kernel<<<NBLOCKS, TPB, 0, stream>>>(pos, eps, sig, ei,
                                                out + 1, ws, n_edges, NBLOCKS);

    // 3) WMMA-based final reduction of the 11 scalars (single wave)
    lj_reduce_kernel<<<1, 32, 0, stream>>>(ws, out, NBLOCKS, n_atoms);
}